// RLLoss_5617817224041
// MI455X (gfx1250) — compile-verified
//
#include <hip/hip_runtime.h>
#include <math.h>

#define HH 512
#define WW2 512
#define HW (512*512)          // 2^18
#define NB 16
#define NB0 8
#define MM 2048
#define BIGN ((size_t)NB*HW)

typedef __attribute__((ext_vector_type(2))) float v2f;
typedef __attribute__((ext_vector_type(8))) float v8f;

// ---------------- block reduction helpers ----------------
__device__ __forceinline__ float bredMax(float v, float* s) {
    int t = threadIdx.x, n = blockDim.x;
    s[t] = v; __syncthreads();
    for (int off = n >> 1; off > 0; off >>= 1) {
        if (t < off) s[t] = fmaxf(s[t], s[t + off]);
        __syncthreads();
    }
    float r = s[0]; __syncthreads();
    return r;
}
__device__ __forceinline__ float bredSum(float v, float* s) {
    int t = threadIdx.x, n = blockDim.x;
    s[t] = v; __syncthreads();
    for (int off = n >> 1; off > 0; off >>= 1) {
        if (t < off) s[t] += s[t + off];
        __syncthreads();
    }
    float r = s[0]; __syncthreads();
    return r;
}

// ---------------- constants (gauss 25, coverage 51) ----------------
__global__ void initK(float* g25, float* w51) {
    if (threadIdx.x == 0 && blockIdx.x == 0) {
        float s = 0.f;
        for (int i = 0; i < 25; ++i) {
            float x = -1.f + 2.f * (float)i / 24.f;
            float g = expf(-2.f * x * x);
            g25[i] = g; s += g;
        }
        for (int i = 0; i < 25; ++i) g25[i] /= s;
        for (int i = 0; i < 51; ++i) {
            float x = -2.f + 4.f * (float)i / 50.f;
            w51[i] = expf(-x * x);
        }
    }
}

// ---------------- per-map softmax over 262144 ----------------
__global__ void __launch_bounds__(1024) softmaxK(const float* __restrict__ x, float* __restrict__ p) {
    __shared__ float red[1024];
    int b = blockIdx.x, t = threadIdx.x;
    const float* xb = x + (size_t)b * HW;
    float* pb = p + (size_t)b * HW;
    float mloc = -3.0e38f;
    for (int i = t; i < HW; i += 1024) mloc = fmaxf(mloc, xb[i]);
    float mx = bredMax(mloc, red);
    float sloc = 0.f;
    for (int i = t; i < HW; i += 1024) { float e = expf(xb[i] - mx); pb[i] = e; sloc += e; }
    float s = bredSum(sloc, red);
    float inv = 1.f / s;
    for (int i = t; i < HW; i += 1024) pb[i] *= inv;
}

// ---------------- separable convs (zero pad), with optional affine on input ----------------
__global__ void hconvK(const float* __restrict__ in, float* __restrict__ out,
                       const float* __restrict__ kern, int klen, float scale, float bias) {
    int idx = blockIdx.x * 256 + threadIdx.x;
    if (idx >= (int)(NB * HW)) return;
    int b = idx >> 18, rem = idx & (HW - 1);
    int h = rem >> 9, w = rem & 511;
    int r = klen >> 1;
    const float* row = in + ((size_t)b << 18) + ((size_t)h << 9);
    float acc = 0.f;
    for (int t2 = 0; t2 < klen; ++t2) {
        int wc = w + t2 - r;
        if (wc >= 0 && wc < 512) acc += kern[t2] * (row[wc] * scale + bias);
    }
    out[idx] = acc;
}
__global__ void vconvK(const float* __restrict__ in, float* __restrict__ out,
                       const float* __restrict__ kern, int klen) {
    int idx = blockIdx.x * 256 + threadIdx.x;
    if (idx >= (int)(NB * HW)) return;
    int b = idx >> 18, rem = idx & (HW - 1);
    int h = rem >> 9, w = rem & 511;
    int r = klen >> 1;
    const float* mb = in + ((size_t)b << 18);
    float acc = 0.f;
    for (int t2 = 0; t2 < klen; ++t2) {
        int hc = h + t2 - r;
        if (hc >= 0 && hc < 512) acc += kern[t2] * mb[(hc << 9) + w];
    }
    out[idx] = acc;
}

// ---------------- matchability: KL( p || softmax(log(ss+1e-8)) ) ----------------
__global__ void __launch_bounds__(1024) matchK(const float* __restrict__ ss, const float* __restrict__ sd,
                                               float* __restrict__ matchv) {
    __shared__ float red[1024];
    int b = blockIdx.x, t = threadIdx.x;
    const float* sb = ss + ((size_t)b << 18);
    const float* db = sd + ((size_t)b << 18);
    float mloc = -3.0e38f, sdloc = 0.f;
    for (int i = t; i < HW; i += 1024) {
        int w = i & 511;
        float s = (w >= 12 && w <= 499) ? sb[i] : 0.f;   // asymmetric-padding "bug": zero cols
        mloc = fmaxf(mloc, logf(s + 1e-8f));
        sdloc += db[i];
    }
    float maxL  = bredMax(mloc, red);
    float sumSd = bredSum(sdloc, red);
    float eloc = 0.f;
    for (int i = t; i < HW; i += 1024) {
        int w = i & 511;
        float s = (w >= 12 && w <= 499) ? sb[i] : 0.f;
        eloc += expf(logf(s + 1e-8f) - maxL);
    }
    float sumE = bredSum(eloc, red);
    float logZ = maxL + logf(sumE);
    float acc = 0.f;
    for (int i = t; i < HW; i += 1024) {
        int w = i & 511;
        float s = (w >= 12 && w <= 499) ? sb[i] : 0.f;
        float L = logf(s + 1e-8f);
        float p = db[i] / sumSd;
        if (p > 0.f) acc += p * (logf(fmaxf(p, 1e-30f)) - (L - logZ));
    }
    float tot = bredSum(acc, red);
    if (t == 0) matchv[b] = tot;
}

// ---------------- coverage reweight + NMS ----------------
__global__ void sm2K(const float* __restrict__ kp, const float* __restrict__ ld, float* __restrict__ out) {
    int idx = blockIdx.x * 256 + threadIdx.x;
    if (idx >= (int)(NB * HW)) return;
    out[idx] = kp[idx] * powf(ld[idx] + 1e-8f, -0.5f);
}
__global__ void nmsK(const float* __restrict__ sm2, unsigned* __restrict__ keys) {
    int idx = blockIdx.x * 256 + threadIdx.x;
    if (idx >= (int)(NB * HW)) return;
    int b = idx >> 18, rem = idx & (HW - 1);
    int h = rem >> 9, w = rem & 511;
    const float* mb = sm2 + ((size_t)b << 18);
    float c = mb[rem], mx = c;
    #pragma unroll
    for (int dy = -2; dy <= 2; ++dy)
        #pragma unroll
        for (int dx = -2; dx <= 2; ++dx) {
            int hh = h + dy, wc = w + dx;
            if (hh >= 0 && hh < 512 && wc >= 0 && wc < 512) mx = fmaxf(mx, mb[(hh << 9) + wc]);
        }
    keys[idx] = (c == mx) ? __float_as_uint(c) : 0u;   // values >= 0 -> uint order == float order
}

// ---------------- exact top-2048: 4-level radix select ----------------
__global__ void zeroHistK(unsigned* hist) {
    int i = blockIdx.x * 256 + threadIdx.x;
    if (i < NB * 256) hist[i] = 0u;
}
__global__ void histK(const unsigned* __restrict__ keys, unsigned* __restrict__ hist,
                      const unsigned* __restrict__ pfx, int level) {
    int idx = blockIdx.x * 256 + threadIdx.x;
    if (idx >= (int)(NB * HW)) return;
    int b = idx >> 18;
    unsigned key = keys[idx];
    int shift = 24 - 8 * level;
    bool ok = (level == 0) || ((key >> (shift + 8)) == pfx[b]);
    if (ok) atomicAdd(&hist[(b << 8) + ((key >> shift) & 255u)], 1u);
}
__global__ void selK(unsigned* hist, unsigned* pfx, unsigned* remk, int level) {
    int b = threadIdx.x;
    if (b >= NB) return;
    unsigned need = (level == 0) ? (unsigned)MM : remk[b];
    unsigned cum = 0; int chosen = 0;
    for (int u = 255; u >= 0; --u) {
        unsigned c = hist[(b << 8) + u];
        if (cum + c >= need) { chosen = u; break; }
        cum += c;
    }
    pfx[b]  = (level == 0) ? (unsigned)chosen : ((pfx[b] << 8) | (unsigned)chosen);
    remk[b] = need - cum;
    for (int u = 0; u < 256; ++u) hist[(b << 8) + u] = 0u;   // ready for next level
}
// deterministic slot assignment (index-ordered ties, like lax.top_k)
__global__ void __launch_bounds__(1024) collectK(const unsigned* __restrict__ keys,
                                                 const unsigned* __restrict__ pfx,
                                                 const unsigned* __restrict__ remk,
                                                 float* __restrict__ kx, float* __restrict__ ky) {
    __shared__ unsigned sb[1024];
    int b = blockIdx.x, t = threadIdx.x;
    unsigned T = pfx[b], R = remk[b];
    unsigned countG = (unsigned)MM - R;
    unsigned runG = 0, runE = 0;
    const unsigned* kb = keys + ((size_t)b << 18);
    for (int chunk = 0; chunk < HW / 1024; ++chunk) {
        int i = chunk * 1024 + t;
        unsigned key = kb[i];
        unsigned g = key > T ? 1u : 0u;
        unsigned e = key == T ? 1u : 0u;
        unsigned v = g | (e << 16);
        sb[t] = v; __syncthreads();
        for (int off = 1; off < 1024; off <<= 1) {
            unsigned add = (t >= off) ? sb[t - off] : 0u;
            __syncthreads();
            sb[t] += add;
            __syncthreads();
        }
        unsigned exc = sb[t] - v;
        int slot = -1;
        if (g) slot = (int)(runG + (exc & 0xFFFFu));
        else if (e) {
            unsigned eidx = runE + (exc >> 16);
            if (eidx < R) slot = (int)(countG + eidx);
        }
        if (slot >= 0 && slot < MM) {
            int col = i & 511, row = i >> 9;
            kx[b * MM + slot] = ((float)col + 0.5f) * (2.0f / 512.f) - 1.0f;
            ky[b * MM + slot] = ((float)row + 0.5f) * (2.0f / 512.f) - 1.0f;
        }
        unsigned tot = sb[1023];
        __syncthreads();
        runG += tot & 0xFFFFu; runE += tot >> 16;
    }
}

// ---------------- bilinear sampling (align_corners=False, zero pad) ----------------
__device__ __forceinline__ float bsample(const float* __restrict__ base, int stride, float nx, float ny) {
    float x = (nx + 1.0f) * 0.5f * 512.f - 0.5f;
    float y = (ny + 1.0f) * 0.5f * 512.f - 0.5f;
    float x0f = floorf(x), y0f = floorf(y);
    int x0 = (int)x0f, y0 = (int)y0f;
    float wx = x - x0f, wy = y - y0f;
    float acc = 0.f;
    #pragma unroll
    for (int dy = 0; dy < 2; ++dy)
        #pragma unroll
        for (int dx = 0; dx < 2; ++dx) {
            int yi = y0 + dy, xi = x0 + dx;
            if (yi >= 0 && yi < 512 && xi >= 0 && xi < 512) {
                float wgt = (dy ? wy : 1.f - wy) * (dx ? wx : 1.f - wx);
                acc += base[((size_t)yi * 512 + xi) * stride] * wgt;
            }
        }
    return acc;
}
__global__ void sampleK(const float* __restrict__ scoremap,
                        const float* __restrict__ wA, const float* __restrict__ wB,
                        const float* __restrict__ vA, const float* __restrict__ vB,
                        const float* __restrict__ kx, const float* __restrict__ ky,
                        float* __restrict__ qX, float* __restrict__ qY,
                        float* __restrict__ leg, float* __restrict__ sl) {
    int idx = blockIdx.x * 256 + threadIdx.x;
    if (idx >= 2 * NB0 * MM) return;
    int m = idx & (MM - 1);
    int b = (idx >> 11) & 7;
    int dir = idx >> 14;
    int kmap = dir ? (NB0 + b) : b;
    float nx = kx[kmap * MM + m], ny = ky[kmap * MM + m];
    const float* warp = (dir ? wB : wA) + (size_t)b * HW * 4;
    qX[idx] = bsample(warp + 2, 4, nx, ny);             // channel 2
    qY[idx] = bsample(warp + 3, 4, nx, ny);             // channel 3
    const float* vv = (dir ? vB : vA) + (size_t)b * HW;
    leg[idx] = bsample(vv, 1, nx, ny) > 0.f ? 1.f : 0.f;
    sl[idx]  = bsample(scoremap + (size_t)kmap * HW, 1, nx, ny);
}

// ---------------- WMMA pairwise-distance min -> reward ----------------
// D = A(16x4) * B(4x16), A row = [qx, qy, |q|^2, 1], B col = [-2px, -2py, 1, |p|^2] -> dist^2
__global__ void __launch_bounds__(256) wmmaDistK(const float* __restrict__ qX, const float* __restrict__ qY,
                                                 const float* __restrict__ kx, const float* __restrict__ ky,
                                                 float* __restrict__ rew) {
    int lane = threadIdx.x & 31;
    int wave = threadIdx.x >> 5;
    int qt = blockIdx.x * 8 + wave;         // query tile 0..127
    int b = blockIdx.y, dir = blockIdx.z;
    int d = dir * NB0 + b;
    int tmap = dir ? b : (NB0 + b);         // targets: kpts_B for dir A, kpts_A for dir B
    int l16 = lane & 15, half = lane >> 4;

    int qm = qt * 16 + l16;
    float qx = qX[d * MM + qm], qy = qY[d * MM + qm];
    // A-matrix 16x4 f32 layout: lanes 0-15: K=0,1 ; lanes 16-31: K=2,3
    v2f a;
    a[0] = half ? (qx * qx + qy * qy) : qx;
    a[1] = half ? 1.0f : qy;

    float rmin[8];
    #pragma unroll
    for (int r = 0; r < 8; ++r) rmin[r] = 3.0e38f;

    const float* tx = kx + tmap * MM;
    const float* ty = ky + tmap * MM;
    v8f czero = {};
    #pragma unroll 4
    for (int nt = 0; nt < 128; ++nt) {
        float px = tx[nt * 16 + l16], py = ty[nt * 16 + l16];
        // B-matrix 4x16 f32: VGPR0: K=0 (lanes 0-15) / K=2 (lanes 16-31); VGPR1: K=1 / K=3
        v2f bb;
        bb[0] = half ? 1.0f : (-2.f * px);
        bb[1] = half ? (px * px + py * py) : (-2.f * py);
        v8f dsq = __builtin_amdgcn_wmma_f32_16x16x4_f32(false, a, false, bb,
                                                        (short)0, czero, false, false);
        #pragma unroll
        for (int r = 0; r < 8; ++r) rmin[r] = fminf(rmin[r], dsq[r]);
    }
    // per-row min: rows r live in lanes 0-15, rows r+8 in lanes 16-31 (C/D layout)
    #pragma unroll
    for (int off = 1; off < 16; off <<= 1) {
        #pragma unroll
        for (int r = 0; r < 8; ++r)
            rmin[r] = fminf(rmin[r], __shfl_xor(rmin[r], off, 32));
    }
    if (l16 == 0) {
        #pragma unroll
        for (int r = 0; r < 8; ++r) {
            int row = qt * 16 + half * 8 + r;
            rew[d * MM + row] = expf(-100.f * sqrtf(rmin[r] + 1e-12f));
        }
    }
}

// ---------------- masked log-softmax REINFORCE term ----------------
__global__ void __launch_bounds__(256) sparseK(const float* __restrict__ sl, const float* __restrict__ leg,
                                               const float* __restrict__ rew, float* __restrict__ part) {
    __shared__ float red[256];
    int d = blockIdx.x, t = threadIdx.x;
    float mloc = -3.0e38f;
    for (int m = t; m < MM; m += 256) {
        float ml = leg[d * MM + m] > 0.f ? sl[d * MM + m] : -1e9f;
        mloc = fmaxf(mloc, ml);
    }
    float mx = bredMax(mloc, red);
    float eloc = 0.f;
    for (int m = t; m < MM; m += 256) {
        float ml = leg[d * MM + m] > 0.f ? sl[d * MM + m] : -1e9f;
        eloc += expf(ml - mx);
    }
    float se = bredSum(eloc, red);
    float lz = logf(se);
    float acc = 0.f;
    for (int m = t; m < MM; m += 256) {
        if (leg[d * MM + m] > 0.f) {
            float lp = sl[d * MM + m] - mx - lz;
            acc += rew[d * MM + m] * lp;
        }
    }
    float s = bredSum(acc, red);
    if (t == 0) part[d] = -s;
}

__global__ void finalK(const float* __restrict__ part, const float* __restrict__ matchv, float* out) {
    if (threadIdx.x == 0 && blockIdx.x == 0) {
        float s = 0.f;
        for (int i = 0; i < 16; ++i) s += part[i];
        float ml = 0.f;
        for (int i = 0; i < 16; ++i) ml += matchv[i];
        out[0] = s + 1.0f * (ml / 16.f);   // REG_W = 1.0
    }
}

extern "C" void kernel_launch(void* const* d_in, const int* in_sizes, int n_in,
                              void* d_out, int out_size, void* d_ws, size_t ws_size,
                              hipStream_t stream) {
    const float* scoremap = (const float*)d_in[0];
    const float* wA = (const float*)d_in[1];
    const float* wB = (const float*)d_in[2];
    const float* vA = (const float*)d_in[3];
    const float* vB = (const float*)d_in[4];
    const float* hd = (const float*)d_in[5];
    float* out = (float*)d_out;

    // workspace carve (~68 MB assumed available)
    float* kp     = (float*)d_ws;          // kp_p, 16*262144
    float* buf1   = kp   + BIGN;           // conv scratch / keys
    float* buf2   = buf1 + BIGN;           // ss / ld
    float* buf3   = buf2 + BIGN;           // sd / sm2
    float* kptsX  = buf3 + BIGN;           // 16*2048
    float* kptsY  = kptsX + NB * MM;
    float* qX     = kptsY + NB * MM;       // 16*2048 (dir*8+b)
    float* qY     = qX + NB * MM;
    float* leg    = qY + NB * MM;
    float* sl     = leg + NB * MM;
    float* rew    = sl + NB * MM;
    float* matchv = rew + NB * MM;         // 16
    float* part   = matchv + NB;           // 16
    float* g25    = part + NB;             // 32
    float* w51    = g25 + 32;              // 64
    unsigned* hist = (unsigned*)(w51 + 64);// 16*256
    unsigned* pfx  = hist + NB * 256;      // 16
    unsigned* remk = pfx + NB;             // 16
    unsigned* keys = (unsigned*)buf1;

    const int nbBlocks = (int)(NB * HW) / 256;   // 16384

    initK<<<1, 64, 0, stream>>>(g25, w51);
    softmaxK<<<NB, 1024, 0, stream>>>(scoremap, kp);

    // ss = (masked-later) separable 25-tap blur of kp_p
    hconvK<<<nbBlocks, 256, 0, stream>>>(kp, buf1, g25, 25, 1.f, 0.f);
    vconvK<<<nbBlocks, 256, 0, stream>>>(buf1, buf2, g25, 25);
    // sd = separable blur of has_depth
    hconvK<<<nbBlocks, 256, 0, stream>>>(hd, buf1, g25, 25, 1.f, 0.f);
    vconvK<<<nbBlocks, 256, 0, stream>>>(buf1, buf3, g25, 25);
    matchK<<<NB, 1024, 0, stream>>>(buf2, buf3, matchv);

    // coverage blur: input (kp + 1e-6) * 1e4 == kp*1e4 + 1e-2
    hconvK<<<nbBlocks, 256, 0, stream>>>(kp, buf1, w51, 51, 1e4f, 1e-2f);
    vconvK<<<nbBlocks, 256, 0, stream>>>(buf1, buf2, w51, 51);
    sm2K<<<nbBlocks, 256, 0, stream>>>(kp, buf2, buf3);
    nmsK<<<nbBlocks, 256, 0, stream>>>(buf3, keys);

    zeroHistK<<<16, 256, 0, stream>>>(hist);
    for (int level = 0; level < 4; ++level) {
        histK<<<nbBlocks, 256, 0, stream>>>(keys, hist, pfx, level);
        selK<<<1, 16, 0, stream>>>(hist, pfx, remk, level);
    }
    collectK<<<NB, 1024, 0, stream>>>(keys, pfx, remk, kptsX, kptsY);

    sampleK<<<(2 * NB0 * MM) / 256, 256, 0, stream>>>(scoremap, wA, wB, vA, vB,
                                                      kptsX, kptsY, qX, qY, leg, sl);
    wmmaDistK<<<dim3(16, 8, 2), 256, 0, stream>>>(qX, qY, kptsX, kptsY, rew);
    sparseK<<<16, 256, 0, stream>>>(sl, leg, rew, part);
    finalK<<<1, 1, 0, stream>>>(part, matchv, out);
}